// PureLRUR_5763846111453
// MI455X (gfx1250) — compile-verified
//
#include <hip/hip_runtime.h>
#include <cstddef>
#include <cstdint>

// PureLRU for MI455X (gfx1250, wave32, WMMA + async LDS staging).
// Phase A: on-device parameter synthesis (scalar kernels + single-WG solvers).
// Phase B: chunked parallel linear scan + big GEMMs, all fp32 WMMA 16x16x4.
// d_out layout: [outputs 16.7M f32][states 16.7M f32]; outputs region doubles
// as staging for F = x @ B^T (t-major) until the final pass overwrites it.
// d_ws usage: ~15 MB of f32 scratch (see offsets in kernel_launch).

#define NDIM   256
#define TSEQ   2048
#define BATCH  32
#define CHUNK  16
#define NCHUNK 128          // TSEQ / CHUNK
#define SSTR   268          // LDS row stride (268 mod 64 = 12, coprime period 16 -> conflict-free)

typedef __attribute__((ext_vector_type(2))) float v2f;
typedef __attribute__((ext_vector_type(8))) float v8f;

__device__ __forceinline__ v8f wmma4(v2f a, v2f b, v8f c) {
  // D(16x16,f32) = A(16x4,f32) x B(4x16,f32) + C
  return __builtin_amdgcn_wmma_f32_16x16x4_f32(false, a, false, b, (short)0, c, false, false);
}

// Async memory->LDS copy of 16 bytes per lane (CDNA5 GLOBAL_LOAD_ASYNC_TO_LDS,
// tracked by ASYNCcnt). LDS offset = low 32 bits of the flat LDS pointer.
__device__ __forceinline__ void async_lds_b128(float* lds_dst, const float* gsrc) {
  uint32_t loff = (uint32_t)(uintptr_t)lds_dst;
  uint64_t ga   = (uint64_t)(uintptr_t)gsrc;
  asm volatile("global_load_async_to_lds_b128 %0, %1, off"
               :: "v"(loff), "v"(ga) : "memory");
}
__device__ __forceinline__ void wait_async0() {
  asm volatile("s_wait_asynccnt 0x0" ::: "memory");
}

// Stage a 256x256 f32 matrix into LDS with padded row stride SSTR using the
// async path: 16384 B128 transfers, 64 per thread (ASYNCcnt auto-throttles).
__device__ __forceinline__ void stage_matrix_async(float* __restrict__ Alds,
                                                   const float* __restrict__ src) {
  const int tid = threadIdx.x;      // 256 threads, all lanes active
  for (int i = tid; i < 16384; i += 256) {
    int row = i >> 6, q = (i & 63) * 4;          // 64 float4 per 256-float row
    async_lds_b128(Alds + row * SSTR + q, src + row * NDIM + q);
  }
  wait_async0();
}

// ---------------------------------------------------------------------------
// Phase A kernels (setup: tiny work, scalar is fine)
// ---------------------------------------------------------------------------

// Unpack packed tril/triu params into dense X11L, X22L (lower) and SK (skew).
__global__ void unpack_params(float* __restrict__ X11L, float* __restrict__ X22L,
                              float* __restrict__ SK,
                              const float* __restrict__ p11, const float* __restrict__ p22,
                              const float* __restrict__ psk) {
  int idx = blockIdx.x * 256 + threadIdx.x;       // 65536 threads
  int i = idx >> 8, j = idx & 255;
  float l11 = 0.f, l22 = 0.f;
  if (j <= i) { int o = i * (i + 1) / 2 + j; l11 = p11[o]; l22 = p22[o]; }
  X11L[idx] = l11;
  X22L[idx] = l22;
  float s = 0.f;
  if (j > i)      { int o = i * 255 - i * (i - 1) / 2 + (j - i - 1); s =  psk[o]; }
  else if (i > j) { int o = j * 255 - j * (j - 1) / 2 + (i - j - 1); s = -psk[o]; }
  SK[idx] = s;
}

// out = betaH*out + aH*(aD?*aD:1) * op(A)op(B) + eyeH*(eyeD?*eyeD:1)*I
__global__ void sgemm256(float* __restrict__ out, const float* __restrict__ A,
                         const float* __restrict__ B, int ta, int tb,
                         float aH, const float* aD, float betaH,
                         float eyeH, const float* eyeD) {
  int idx = blockIdx.x * 256 + threadIdx.x;
  int i = idx >> 8, j = idx & 255;
  float acc = 0.f;
#pragma unroll 4
  for (int k = 0; k < NDIM; ++k) {
    float av = ta ? A[k * NDIM + i] : A[i * NDIM + k];
    float bv = tb ? B[j * NDIM + k] : B[k * NDIM + j];
    acc = fmaf(av, bv, acc);
  }
  float alpha = aH * (aD ? *aD : 1.f);
  float eye   = eyeH * (eyeD ? *eyeD : 1.f);
  float prev  = (betaH != 0.f) ? out[idx] : 0.f;
  out[idx] = betaH * prev + alpha * acc + ((i == j) ? eye : 0.f);
}

// out = aH*(aD?*aD:1)*X + eyeH*(eyeD?*eyeD:1)*I
__global__ void axpby_eye(float* __restrict__ out, const float* __restrict__ X,
                          float aH, const float* aD, float eyeH, const float* eyeD) {
  int idx = blockIdx.x * 256 + threadIdx.x;
  int i = idx >> 8, j = idx & 255;
  float a = aH * (aD ? *aD : 1.f);
  float e = eyeH * (eyeD ? *eyeD : 1.f);
  out[idx] = a * X[idx] + ((i == j) ? e : 0.f);
}

// Power iteration: lam = ||Z||_2 (Z SPD); scal = {beta, sqrt(beta), gamma^2}
__global__ void powernorm(const float* __restrict__ Z, const float* __restrict__ gamma,
                          const float* __restrict__ alpha, float* __restrict__ scal) {
  __shared__ float v[NDIM];
  __shared__ float red[NDIM];
  int i = threadIdx.x;
  v[i] = 1.f;
  __syncthreads();
  float lam = 1.f;
  for (int it = 0; it < 96; ++it) {
    float w = 0.f;
    const float* row = Z + i * NDIM;
#pragma unroll 4
    for (int k = 0; k < NDIM; ++k) w = fmaf(row[k], v[k], w);
    red[i] = w * w;
    __syncthreads();
    for (int s = 128; s > 0; s >>= 1) {
      if (i < s) red[i] += red[i + s];
      __syncthreads();
    }
    lam = sqrtf(red[0]);
    __syncthreads();
    v[i] = w / lam;
    __syncthreads();
  }
  if (i == 0) {
    float g = gamma[0], a = alpha[0];
    float sig  = 1.f / (1.f + expf(-a));
    float beta = g * g * sig / lam;
    scal[0] = beta;
    scal[1] = sqrtf(beta);
    scal[2] = g * g;
  }
}

// Single-WG Gauss-Jordan: solve M X = R (256x256, no pivoting; inputs are
// identity-dominated / definite). AUG = 256x512 scratch in global ws.
__global__ void gauss_solve(float* __restrict__ AUG, const float* __restrict__ M, int ta,
                            const float* __restrict__ R, int tb, float* __restrict__ X) {
  int tid = threadIdx.x;            // 1024 threads
  int row = tid >> 2;               // each thread owns (row, 128-col segment)
  int seg = tid & 3;
  for (int idx = tid; idx < 256 * 512; idx += 1024) {
    int i = idx >> 9, j = idx & 511;
    float val;
    if (j < 256) val = ta ? M[j * NDIM + i] : M[i * NDIM + j];
    else { int jj = j - 256; val = tb ? R[jj * NDIM + i] : R[i * NDIM + jj]; }
    AUG[idx] = val;
  }
  __threadfence(); __syncthreads();
  for (int k = 0; k < 256; ++k) {
    float piv = AUG[k * 512 + k];
    float f   = AUG[row * 512 + k] / piv;
    __syncthreads();                      // all reads done before updates
    if (row != k) {
      const float* rk = AUG + k * 512 + seg * 128;
      float*       ri = AUG + row * 512 + seg * 128;
#pragma unroll 4
      for (int j = 0; j < 128; ++j) ri[j] = fmaf(-f, rk[j], ri[j]);
    }
    __threadfence(); __syncthreads();
    if (row == k) {
      float inv = 1.f / piv;
      float* rk = AUG + k * 512 + seg * 128;
      for (int j = 0; j < 128; ++j) rk[j] *= inv;
    }
    __threadfence(); __syncthreads();
  }
  for (int idx = tid; idx < NDIM * NDIM; idx += 1024) {
    int i = idx >> 8, j = idx & 255;
    X[idx] = AUG[i * 512 + 256 + j];
  }
}

// Single-WG Cholesky: L = chol(alphaH*In + Add). Mbuf = global scratch.
__global__ void cholesky256(float* __restrict__ L, const float* __restrict__ In,
                            float alphaH, const float* __restrict__ Add,
                            float* __restrict__ Mbuf) {
  __shared__ float col[NDIM];
  int tid = threadIdx.x;            // 256 threads, thread owns row tid
  for (int idx = tid; idx < NDIM * NDIM; idx += 256) {
    Mbuf[idx] = alphaH * In[idx] + (Add ? Add[idx] : 0.f);
    L[idx] = 0.f;
  }
  __threadfence(); __syncthreads();
  for (int k = 0; k < 256; ++k) {
    if (tid >= k) {
      float d = sqrtf(Mbuf[k * NDIM + k]);
      col[tid] = (tid == k) ? d : Mbuf[tid * NDIM + k] / d;
    }
    __syncthreads();
    if (tid > k) {
      float lik = col[tid];
      for (int j = k + 1; j <= tid; ++j)
        Mbuf[tid * NDIM + j] = fmaf(-lik, col[j], Mbuf[tid * NDIM + j]);
    }
    if (tid >= k) L[tid * NDIM + k] = col[tid];
    __threadfence(); __syncthreads();
  }
}

// ---------------------------------------------------------------------------
// Phase B: WMMA kernels
// ---------------------------------------------------------------------------

// One recurrence step: S(32x256) = S @ A_lds^T + F_t, done by 8 waves.
// Wave w owns output tiles (mi in {0,1}) x (ni in {2w, 2w+1}).
__device__ __forceinline__ void scan_step(const float* __restrict__ Alds,
                                          float* __restrict__ S,
                                          const float* __restrict__ Ft,
                                          float* __restrict__ states, int t) {
  const int tid  = threadIdx.x;
  const int wave = tid >> 5, lane = tid & 31, lo = lane & 15, hi = lane >> 4;
  const int nc0 = (wave * 2 + 0) * 16 + lo;
  const int nc1 = (wave * 2 + 1) * 16 + lo;
  v8f a00, a01, a10, a11;
#pragma unroll
  for (int r = 0; r < 8; ++r) {             // C layout: vgpr r <-> M = r + 8*hi
    int m = r + 8 * hi;
    a00[r] = Ft[m * NDIM + nc0];
    a01[r] = Ft[m * NDIM + nc1];
    a10[r] = Ft[(16 + m) * NDIM + nc0];
    a11[r] = Ft[(16 + m) * NDIM + nc1];
  }
#pragma unroll 8
  for (int k = 0; k < 64; ++k) {
    int k0 = 4 * k + 2 * hi;
    v2f af0 = *(const v2f*)(S + lo * SSTR + k0);              // A-frag: M=lo
    v2f af1 = *(const v2f*)(S + (16 + lo) * SSTR + k0);
    v2f bf0 = *(const v2f*)(Alds + nc0 * SSTR + k0);          // B-frag: N=lo, B[k][n]=A[n][k]
    v2f bf1 = *(const v2f*)(Alds + nc1 * SSTR + k0);
    a00 = wmma4(af0, bf0, a00);
    a01 = wmma4(af0, bf1, a01);
    a10 = wmma4(af1, bf0, a10);
    a11 = wmma4(af1, bf1, a11);
  }
  __syncthreads();                           // everyone done reading S
#pragma unroll
  for (int r = 0; r < 8; ++r) {
    int m = r + 8 * hi;
    S[m * SSTR + nc0]        = a00[r];
    S[m * SSTR + nc1]        = a01[r];
    S[(16 + m) * SSTR + nc0] = a10[r];
    S[(16 + m) * SSTR + nc1] = a11[r];
    if (states) {                            // states[b][t][n], b = m row
      states[(size_t)m * (TSEQ * NDIM) + (size_t)t * NDIM + nc0]        = a00[r];
      states[(size_t)m * (TSEQ * NDIM) + (size_t)t * NDIM + nc1]        = a01[r];
      states[(size_t)(16 + m) * (TSEQ * NDIM) + (size_t)t * NDIM + nc0] = a10[r];
      states[(size_t)(16 + m) * (TSEQ * NDIM) + (size_t)t * NDIM + nc1] = a11[r];
    }
  }
  __syncthreads();
}

// mode 0: local scan from zero, emit chunk-end -> yend.  mode 1: scan from
// carry[c], emit every state.  LDS: A (256x268) + S (32x268) ~ 302 KB.
__global__ void scan_chunk(const float* __restrict__ Amat, const float* __restrict__ F,
                           const float* __restrict__ carry, float* __restrict__ yend,
                           float* __restrict__ states, int mode) {
  extern __shared__ float lds[];
  float* Alds = lds;
  float* S    = lds + NDIM * SSTR;
  const int tid = threadIdx.x;
  const int c   = blockIdx.x;
  stage_matrix_async(Alds, Amat);            // async memory->LDS (ASYNCcnt)
  for (int i = tid; i < BATCH * NDIM; i += 256)
    S[(i >> 8) * SSTR + (i & 255)] = (mode == 1) ? carry[c * BATCH * NDIM + i] : 0.f;
  __syncthreads();
  for (int tl = 0; tl < CHUNK; ++tl) {
    int t = c * CHUNK + tl;
    if (tl + 1 < CHUNK)                      // prefetch next 32KB F tile (128B/thread)
      __builtin_prefetch(F + (size_t)(t + 1) * BATCH * NDIM + tid * 32, 0, 1);
    scan_step(Alds, S, F + (size_t)t * BATCH * NDIM, (mode == 1) ? states : nullptr, t);
  }
  if (mode == 0) {
    for (int i = tid; i < BATCH * NDIM; i += 256)
      yend[c * BATCH * NDIM + i] = S[(i >> 8) * SSTR + (i & 255)];
  }
}

// Sequential carry combine over 128 chunks with A^CHUNK (single workgroup).
__global__ void scan_combine(const float* __restrict__ A16, const float* __restrict__ yend,
                             float* __restrict__ carry) {
  extern __shared__ float lds[];
  float* Alds = lds;
  float* S    = lds + NDIM * SSTR;
  const int tid = threadIdx.x;
  stage_matrix_async(Alds, A16);
  for (int i = tid; i < BATCH * NDIM; i += 256)
    S[(i >> 8) * SSTR + (i & 255)] = 0.f;
  __syncthreads();
  for (int c = 0; c < NCHUNK; ++c) {
    for (int i = tid; i < BATCH * NDIM; i += 256)
      carry[c * BATCH * NDIM + i] = S[(i >> 8) * SSTR + (i & 255)];
    if (c + 1 < NCHUNK)
      __builtin_prefetch(yend + (size_t)(c + 1) * BATCH * NDIM + tid * 32, 0, 1);
    scan_step(Alds, S, yend + (size_t)c * BATCH * NDIM, nullptr, 0);
  }
}

// out[orow][0:256] (+)= X[r][0:256] @ W^T, W staged in LDS (async).
// remap=1: orow = t*BATCH + b for r = b*TSEQ + t  (builds F t-major).
__global__ void wmma_gemm_nt(float* __restrict__ out, const float* __restrict__ X,
                             const float* __restrict__ W, int accumulate, int remap) {
  extern __shared__ float lds[];
  const int tid = threadIdx.x;
  stage_matrix_async(lds, W);                // async memory->LDS (ASYNCcnt)
  __syncthreads();
  const int wave = tid >> 5, lane = tid & 31, lo = lane & 15, hi = lane >> 4;
  const int row0 = blockIdx.x * 128 + wave * 16;   // M-tile per wave
  for (int half = 0; half < 2; ++half) {           // ni split to bound registers
    v8f acc[8];
#pragma unroll
    for (int j = 0; j < 8; ++j) {
      int n = (half * 8 + j) * 16 + lo;
#pragma unroll
      for (int r = 0; r < 8; ++r) {
        if (accumulate) {
          int m = row0 + r + 8 * hi;
          int orow = remap ? ((m & (TSEQ - 1)) * BATCH + (m >> 11)) : m;
          acc[j][r] = out[(size_t)orow * NDIM + n];
        } else acc[j][r] = 0.f;
      }
    }
#pragma unroll 4
    for (int k = 0; k < 64; ++k) {
      int k0 = 4 * k + 2 * hi;
      v2f af = *(const v2f*)(X + (size_t)(row0 + lo) * NDIM + k0);
#pragma unroll
      for (int j = 0; j < 8; ++j) {
        v2f bf = *(const v2f*)(lds + ((half * 8 + j) * 16 + lo) * SSTR + k0);
        acc[j] = wmma4(af, bf, acc[j]);
      }
    }
#pragma unroll
    for (int j = 0; j < 8; ++j) {
      int n = (half * 8 + j) * 16 + lo;
#pragma unroll
      for (int r = 0; r < 8; ++r) {
        int m = row0 + r + 8 * hi;
        int orow = remap ? ((m & (TSEQ - 1)) * BATCH + (m >> 11)) : m;
        out[(size_t)orow * NDIM + n] = acc[j][r];
      }
    }
  }
}

// ---------------------------------------------------------------------------

extern "C" void kernel_launch(void* const* d_in, const int* in_sizes, int n_in,
                              void* d_out, int out_size, void* d_ws, size_t ws_size,
                              hipStream_t stream) {
  (void)in_sizes; (void)n_in; (void)out_size; (void)ws_size;
  const float* x     = (const float*)d_in[0];
  const float* gamma = (const float*)d_in[1];
  const float* alpha = (const float*)d_in[2];
  const float* p11   = (const float*)d_in[3];
  const float* p22   = (const float*)d_in[4];
  const float* psk   = (const float*)d_in[5];
  const float* X21   = (const float*)d_in[6];
  const float* Cmat  = (const float*)d_in[7];
  const float* Dt    = (const float*)d_in[8];

  float* outp   = (float*)d_out;                         // outputs [B][T][N]
  float* states = outp + (size_t)BATCH * TSEQ * NDIM;    // states  [B][T][N]
  float* Fbuf   = outp;                                  // F staging [T][B][N] (dead before pass 4)

  float* ws = (float*)d_ws;
  const size_t MAT = (size_t)NDIM * NDIM;
  float* SCAL = ws;                 // [0]=beta [1]=sqrt(beta) [2]=gamma^2
  float* base = ws + 16;
  float* X11L = base +  0 * MAT;  float* X22L = base +  1 * MAT;
  float* SK   = base +  2 * MAT;  float* Z    = base +  3 * MAT;
  float* H11  = base +  4 * MAT;  float* TMP  = base +  5 * MAT;
  float* H12  = base +  6 * MAT;  float* V    = base +  7 * MAT;
  float* Y    = base +  8 * MAT;  float* Rm   = base +  9 * MAT;
  float* CR   = base + 10 * MAT;  float* CRH  = base + 11 * MAT;
  float* MM   = base + 12 * MAT;  float* RR   = base + 13 * MAT;
  float* QT   = base + 14 * MAT;  float* Wm   = base + 15 * MAT;
  float* AMAT = base + 16 * MAT;  float* Y2   = base + 17 * MAT;
  float* BMAT = base + 18 * MAT;  float* DMAT = base + 19 * MAT;
  float* P16  = base + 20 * MAT;
  float* AUG  = base + 21 * MAT;  // 2*MAT (GJ scratch; later A^2/A^4)
  float* CHB  = base + 23 * MAT;  // chol scratch; later A^8
  float* YEND = base + 24 * MAT;  // 16*MAT
  float* CARRY= base + 40 * MAT;  // 16*MAT

  dim3 g(256), b(256);
  // ---- Phase A: synthesize A,B,D (C passes through) ----
  unpack_params<<<g, b, 0, stream>>>(X11L, X22L, SK, p11, p22, psk);
  // Z = X21 X21^T + X22 X22^T + Dt^T Dt + I
  sgemm256<<<g, b, 0, stream>>>(Z, X21,  X21,  0, 1, 1.f, nullptr, 0.f, 0.f, nullptr);
  sgemm256<<<g, b, 0, stream>>>(Z, X22L, X22L, 0, 1, 1.f, nullptr, 1.f, 0.f, nullptr);
  sgemm256<<<g, b, 0, stream>>>(Z, Dt,   Dt,   1, 0, 1.f, nullptr, 1.f, 1.f, nullptr);
  powernorm<<<1, b, 0, stream>>>(Z, gamma, alpha, SCAL);
  // H11 = X11 X11^T + C^T C + beta*I
  sgemm256<<<g, b, 0, stream>>>(H11, X11L, X11L, 0, 1, 1.f, nullptr, 0.f, 0.f, nullptr);
  sgemm256<<<g, b, 0, stream>>>(H11, Cmat, Cmat, 1, 0, 1.f, nullptr, 1.f, 1.f, SCAL + 0);
  // H12 = sqrt(beta)*(X11 X21^T + C^T Dt)
  sgemm256<<<g, b, 0, stream>>>(TMP, X11L, X21, 0, 1, 1.f, nullptr, 0.f, 0.f, nullptr);
  sgemm256<<<g, b, 0, stream>>>(TMP, Cmat, Dt,  1, 0, 1.f, nullptr, 1.f, 0.f, nullptr);
  axpby_eye<<<g, b, 0, stream>>>(H12, TMP, 1.f, SCAL + 1, 0.f, nullptr);
  // V = beta*Z - gamma^2 I
  axpby_eye<<<g, b, 0, stream>>>(V, Z, 1.f, SCAL + 0, -1.f, SCAL + 2);
  // Y = inv(V^T) H12^T ; R = H12 Y ; CR=chol(-R) ; CRH=chol(-R+H11)
  gauss_solve<<<1, 1024, 0, stream>>>(AUG, V, 1, H12, 1, Y);
  sgemm256<<<g, b, 0, stream>>>(Rm, H12, Y, 0, 0, 1.f, nullptr, 0.f, 0.f, nullptr);
  cholesky256<<<1, b, 0, stream>>>(CR,  Rm, -1.f, nullptr, CHB);
  cholesky256<<<1, b, 0, stream>>>(CRH, Rm, -1.f, H11,     CHB);
  // Q: (I-Sk) Q^T = (I+Sk)
  axpby_eye<<<g, b, 0, stream>>>(MM, SK, -1.f, nullptr, 1.f, nullptr);
  axpby_eye<<<g, b, 0, stream>>>(RR, SK,  1.f, nullptr, 1.f, nullptr);
  gauss_solve<<<1, 1024, 0, stream>>>(AUG, MM, 0, RR, 0, QT);
  // W = Q CR^T = QT^T CR^T ; solve CRH^T A = W ; solve H12^T Y2 = V^T ; B = A Y2
  sgemm256<<<g, b, 0, stream>>>(Wm, QT, CR, 1, 1, 1.f, nullptr, 0.f, 0.f, nullptr);
  gauss_solve<<<1, 1024, 0, stream>>>(AUG, CRH, 1, Wm, 0, AMAT);
  gauss_solve<<<1, 1024, 0, stream>>>(AUG, H12, 1, V, 1, Y2);
  sgemm256<<<g, b, 0, stream>>>(BMAT, AMAT, Y2, 0, 0, 1.f, nullptr, 0.f, 0.f, nullptr);
  axpby_eye<<<g, b, 0, stream>>>(DMAT, Dt, 1.f, SCAL + 1, 0.f, nullptr);
  // A^16 by repeated squaring (reuse solver/chol scratch)
  sgemm256<<<g, b, 0, stream>>>(AUG,       AMAT,      AMAT,      0, 0, 1.f, nullptr, 0.f, 0.f, nullptr);
  sgemm256<<<g, b, 0, stream>>>(AUG + MAT, AUG,       AUG,       0, 0, 1.f, nullptr, 0.f, 0.f, nullptr);
  sgemm256<<<g, b, 0, stream>>>(CHB,       AUG + MAT, AUG + MAT, 0, 0, 1.f, nullptr, 0.f, 0.f, nullptr);
  sgemm256<<<g, b, 0, stream>>>(P16,       CHB,       CHB,       0, 0, 1.f, nullptr, 0.f, 0.f, nullptr);

  // ---- Phase B: WMMA scan + GEMMs ----
  const uint32_t WLDS = NDIM * SSTR * sizeof(float);            // 274 KB
  const uint32_t SLDS = (NDIM + BATCH) * SSTR * sizeof(float);  // 302 KB (<= 320 KB/WGP)
  // F = x @ B^T, stored t-major in the outputs region
  wmma_gemm_nt<<<512, b, WLDS, stream>>>(Fbuf, x, BMAT, 0, 1);
  // parallel chunked linear scan
  scan_chunk<<<NCHUNK, b, SLDS, stream>>>(AMAT, Fbuf, nullptr, YEND, nullptr, 0);
  scan_combine<<<1, b, SLDS, stream>>>(P16, YEND, CARRY);
  scan_chunk<<<NCHUNK, b, SLDS, stream>>>(AMAT, Fbuf, CARRY, nullptr, states, 1);
  // outputs = states @ C^T + x @ D^T (overwrites dead F)
  wmma_gemm_nt<<<512, b, WLDS, stream>>>(outp, states, Cmat, 0, 0);
  wmma_gemm_nt<<<512, b, WLDS, stream>>>(outp, x, DMAT, 1, 0);
}